// PhaseAwareClassifier_51857435131954
// MI455X (gfx1250) — compile-verified
//
#include <hip/hip_runtime.h>
#include <math.h>

typedef float v2f __attribute__((ext_vector_type(2)));
typedef float v8f __attribute__((ext_vector_type(8)));

#define N_SW   131
#define SPAD   132        // K (s) padded to 33*4
#define TPAD   144        // t padded to 9*16
#define SROWS  144        // out LDS rows (covers padded t writes)
#define NT     9
#define NK     33
#define WAVES  8
#define DECAY  0.15f

// workspace layout (float offsets)
#define WS_ABSMAX 0
#define WS_SCALE  1
#define WS_ENERGY 16                    // 128*10 floats
#define WS_MTR    2048                  // TPAD*SPAD
#define WS_MTI    (WS_MTR + TPAD*SPAD)
#define WS_GAIN   (WS_MTI + TPAD*SPAD)  // TPAD floats

__global__ void pac_init(float* ws) {
  int i = blockIdx.x * 256 + threadIdx.x;
  if (i < 2048) ws[i] = 0.0f;
}

__global__ void pac_absmax(const float* __restrict__ img, float* ws, int n) {
  int i = blockIdx.x * 256 + threadIdx.x;
  float v = 0.0f;
  for (; i < n; i += gridDim.x * 256) v = fmaxf(v, fabsf(img[i]));
  atomicMax(reinterpret_cast<unsigned int*>(ws) + WS_ABSMAX, __float_as_uint(v));
}

// Build rotated connection matrices Mt[t][s] (phase folded in), softplus(gain), and 1/max scale.
__global__ void pac_prep(const float* __restrict__ cr, const float* __restrict__ ci,
                         const float* __restrict__ ph, const float* __restrict__ gw,
                         float* ws) {
  int i = blockIdx.x * 256 + threadIdx.x;
  if (i == 0) {
    float am = __uint_as_float(reinterpret_cast<unsigned int*>(ws)[WS_ABSMAX]);
    ws[WS_SCALE] = (am > 1e-8f) ? (1.0f / am) : 1.0f;
  }
  if (i < TPAD) {
    float g = 0.0f;
    if (i < N_SW) { float x = gw[i]; g = (x > 20.0f) ? x : log1pf(expf(x)); }
    ws[WS_GAIN + i] = g;
  }
  if (i < TPAD * SPAD) {
    int t = i / SPAD, s = i - t * SPAD;
    float mr = 0.0f, mi = 0.0f;
    if (t < N_SW && s < N_SW) {
      float c = cr[s * N_SW + t];
      float d = ci[s * N_SW + t];
      float cp = cosf(ph[t]), sp = sinf(ph[t]);
      mr = c * cp - d * sp;   // gr = Mr^T Or - Mi^T Oi
      mi = c * sp + d * cp;   // gi = Mi^T Or + Mr^T Oi
    }
    ws[WS_MTR + i] = mr;
    ws[WS_MTI + i] = mi;
  }
}

// One wave per (b, L): 16 columns (m=0..15; m>=16 provably stay zero).
// Field kept in VGPRs, out in per-wave LDS slice, conn in shared LDS.
__global__ __launch_bounds__(256, 1) void pac_main(
    const float* __restrict__ img,
    const float* __restrict__ ws,
    float* __restrict__ energy,
    const int* __restrict__ p_nsteps,
    const int* __restrict__ p_injsteps)
{
  extern __shared__ float smem[];
  float* sM    = smem;                      // Mr then Mi: 2*TPAD*SPAD
  float* sGain = smem + 2 * TPAD * SPAD;    // TPAD
  float* sOutA = sGain + TPAD;              // WAVES * 2 * SROWS * 16

  const int tid  = threadIdx.x;
  const int wave = tid >> 5;
  const int lane = tid & 31;
  const int half = lane >> 4;
  const int colN = lane & 15;

  for (int i = tid; i < 2 * TPAD * SPAD; i += 256) sM[i] = ws[WS_MTR + i];
  for (int i = tid; i < TPAD; i += 256) sGain[i] = ws[WS_GAIN + i];

  float* outR = sOutA + wave * (2 * SROWS * 16);
  float* outI = outR + SROWS * 16;
  for (int i = lane; i < 2 * SROWS * 16; i += 32) outR[i] = 0.0f;
  __syncthreads();

  const int p = blockIdx.x * WAVES + wave;      // 0..8191
  const int b = p >> 6;
  const int L = p & 63;
  const float wL = 1.0f - fabsf((float)L - 32.0f) * (1.0f / 64.0f);
  const float injscale = 0.3f * ws[WS_SCALE] * wL * (1.0f - DECAY); // pre-decayed

  // Injection tiles (s<49 rows); C/D layout: lane half h, vgpr e -> row e + 8h
  v8f injv[4];
#pragma unroll
  for (int t = 0; t < 4; ++t) {
#pragma unroll
    for (int e = 0; e < 8; ++e) {
      int s = t * 16 + e + half * 8;
      float v = 0.0f;
      if (s < 49) {
        int pi = s / 7, pj = s - pi * 7;
        int u = colN >> 2, vv = colN & 3;
        v = img[(b * 28 + (pi * 4 + u)) * 28 + (pj * 4 + vv)] * injscale;
      }
      injv[t][e] = v;
    }
  }

  const int nsteps   = __builtin_amdgcn_readfirstlane(p_nsteps[0]);
  const int injsteps = __builtin_amdgcn_readfirstlane(p_injsteps[0]);

  v8f fr[NT], fi[NT];
#pragma unroll
  for (int t = 0; t < NT; ++t) { fr[t] = {}; fi[t] = {}; }

  // A layout (16x4 f32): lanes 0-15 rows M=lane, K pair {0,1}; lanes 16-31 K pair {2,3}
  const float* aRbase = sM + colN * SPAD + 2 * half;
  const float* aIbase = aRbase + TPAD * SPAD;
  // B layout (4x16 f32): lane half h, col=lane&15, K pair {2h, 2h+1}; out stored s-pair interleaved
  const float* bRbase = outR + (half * 16 + colN) * 2;
  const float* bIbase = outI + (half * 16 + colN) * 2;

  for (int step = 0; step < nsteps; ++step) {
    v8f p1[NT], p2[NT], gI[NT];
#pragma unroll
    for (int t = 0; t < NT; ++t) { p1[t] = {}; p2[t] = {}; gI[t] = {}; }

#pragma unroll 1
    for (int kb = 0; kb < NK; ++kb) {
      v2f br = *(const v2f*)(bRbase + kb * 64);
      v2f bi = *(const v2f*)(bIbase + kb * 64);
#pragma unroll
      for (int t = 0; t < NT; ++t) {
        v2f ar = *(const v2f*)(aRbase + t * 16 * SPAD + kb * 4);
        v2f ai = *(const v2f*)(aIbase + t * 16 * SPAD + kb * 4);
        p1[t] = __builtin_amdgcn_wmma_f32_16x16x4_f32(false, ar, false, br, (short)0, p1[t], false, false);
        p2[t] = __builtin_amdgcn_wmma_f32_16x16x4_f32(false, ai, false, bi, (short)0, p2[t], false, false);
        gI[t] = __builtin_amdgcn_wmma_f32_16x16x4_f32(false, ai, false, br, (short)0, gI[t], false, false);
        gI[t] = __builtin_amdgcn_wmma_f32_16x16x4_f32(false, ar, false, bi, (short)0, gI[t], false, false);
      }
    }

    const bool doInj = (step < injsteps);
#pragma unroll
    for (int t = 0; t < NT; ++t) {
      v8f gr  = p1[t] - p2[t];
      v8f f_r = fr[t] * (1.0f - DECAY) + gr * 0.25f;
      v8f f_i = fi[t] * (1.0f - DECAY) + gI[t] * 0.25f;
      if (t < 4 && doInj) f_r = f_r + injv[t];   // (field+inj)*(1-decay) folded
      fr[t] = f_r; fi[t] = f_i;
#pragma unroll
      for (int e = 0; e < 8; ++e) {
        int s = t * 16 + e + half * 8;
        float gs  = sGain[s];                    // 0 for padded rows -> out stays 0
        float xr  = f_r[e], xi = f_i[e];
        float mag = sqrtf(xr * xr + xi * xi + 1e-8f);
        float nm  = tanhf(mag * gs);
        float sc  = nm / (mag + 1e-8f);
        int idx = ((s >> 1) * 16 + colN) * 2 + (s & 1);
        outR[idx] = xr * sc;
        outI[idx] = xi * sc;
      }
    }
  }

  // energy over this wave's 16 columns for s in [121,131)
  if (lane < 10) {
    int s = 121 + lane;
    float e = 0.0f;
#pragma unroll
    for (int c = 0; c < 16; ++c) {
      int idx = ((s >> 1) * 16 + c) * 2 + (s & 1);
      float r_ = outR[idx], i_ = outI[idx];
      e += r_ * r_ + i_ * i_;
    }
    atomicAdd(&energy[b * 10 + lane], e);
  }
}

__global__ void pac_readout(const float* __restrict__ ws,
                            const float* __restrict__ W,
                            const float* __restrict__ bias,
                            float* __restrict__ out) {
  int i = blockIdx.x * 256 + threadIdx.x;
  if (i < 1280) {
    int b = i / 10, c = i - b * 10;
    const float* en = ws + WS_ENERGY + b * 10;
    float acc = bias[c];
#pragma unroll
    for (int k = 0; k < 10; ++k)
      acc += log1pf(en[k] + 1e-8f) * W[c * 10 + k];
    out[i] = acc;
  }
}

extern "C" void kernel_launch(void* const* d_in, const int* in_sizes, int n_in,
                              void* d_out, int out_size, void* d_ws, size_t ws_size,
                              hipStream_t stream) {
  (void)in_sizes; (void)n_in; (void)out_size; (void)ws_size;
  const float* images   = (const float*)d_in[0];
  const float* conn_r   = (const float*)d_in[1];
  const float* conn_i   = (const float*)d_in[2];
  const float* sw_phase = (const float*)d_in[3];
  const float* sw_gain  = (const float*)d_in[4];
  const float* rd_w     = (const float*)d_in[5];
  const float* rd_b     = (const float*)d_in[6];
  const int*   nsteps   = (const int*)d_in[7];
  const int*   injsteps = (const int*)d_in[8];
  float* ws  = (float*)d_ws;
  float* out = (float*)d_out;

  pac_init<<<8, 256, 0, stream>>>(ws);
  pac_absmax<<<64, 256, 0, stream>>>(images, ws, 128 * 28 * 28);
  pac_prep<<<(TPAD * SPAD + 255) / 256, 256, 0, stream>>>(conn_r, conn_i, sw_phase, sw_gain, ws);

  size_t lds = (size_t)(2 * TPAD * SPAD + TPAD + WAVES * 2 * SROWS * 16) * sizeof(float); // ~300 KB
  hipFuncSetAttribute(reinterpret_cast<const void*>(pac_main),
                      hipFuncAttributeMaxDynamicSharedMemorySize, (int)lds);
  pac_main<<<1024, 256, lds, stream>>>(images, ws, ws + WS_ENERGY, nsteps, injsteps);

  pac_readout<<<5, 256, 0, stream>>>(ws, rd_w, rd_b, out);
}